// BlockinnerAttention_61117384622786
// MI455X (gfx1250) — compile-verified
//
#include <hip/hip_runtime.h>
#include <hip/hip_bf16.h>
#include <math.h>

typedef __attribute__((ext_vector_type(2))) float v2f;
typedef __attribute__((ext_vector_type(8))) float v8f;

#define H_    8
#define TQ_   2048
#define TK_   4096
#define DK_   64
#define BS_   64
#define NB_   (TK_ / BS_)
#define CAP_  2048
#define LDSS_ 68   // padded LDS row stride (floats): bank = 4*lane mod 64 -> conflict-free

// ---------------------------------------------------------------------------
// Kernel 1: one workgroup per (head, key-block).
//   Pass A: bin all (query, slot) pairs whose top2 index == this block (LDS atomics).
//   Pass B: per 16-query chunk: S = Q*K^T (f32 WMMA), partial softmax over this
//           block's 64 keys (m, l), O_part = P*V (f32 WMMA). Write partials.
// ---------------------------------------------------------------------------
__global__ __launch_bounds__(64) void bsattn_partial_kernel(
    const float* __restrict__ qg, const float* __restrict__ kg,
    const float* __restrict__ vg, const int* __restrict__ top2,
    float* __restrict__ pO, float* __restrict__ pM, float* __restrict__ pL)
{
  __shared__ int   s_cnt;
  __shared__ int   s_list[CAP_];
  __shared__ float sK[BS_ * LDSS_];
  __shared__ float sV[BS_ * LDSS_];
  __shared__ float sQ[2][16 * LDSS_];
  __shared__ float sP[2][16 * LDSS_];

  const int wg    = blockIdx.x;
  const int h     = wg / NB_;
  const int blk   = wg % NB_;
  const int tid   = threadIdx.x;   // 0..63 (2 waves)
  const int wave  = tid >> 5;
  const int lane  = tid & 31;
  const int lhalf = lane >> 4;     // 0: lanes 0-15, 1: lanes 16-31
  const int lm    = lane & 15;

  if (tid == 0) s_cnt = 0;
  __syncthreads();

  // ---- Pass A: bin queries selecting this block ----
  const int* tp = top2 + (size_t)h * TQ_ * 2;
  for (int i = tid; i < TQ_ * 2; i += 64) {
    if (tp[i] == blk) {
      int pos = atomicAdd(&s_cnt, 1);
      if (pos < CAP_) s_list[pos] = i;     // i = qidx*2 + slot
    }
  }

  // ---- Stage this block's K and V tiles (16 KB each) into padded LDS ----
  const float* kb = kg + ((size_t)h * TK_ + (size_t)blk * BS_) * DK_;
  const float* vb = vg + ((size_t)h * TK_ + (size_t)blk * BS_) * DK_;
  __builtin_prefetch(kb, 0, 1);
  __builtin_prefetch(vb, 0, 1);
  for (int i = tid; i < BS_ * DK_; i += 64) {
    int r = i >> 6, c = i & 63;
    sK[r * LDSS_ + c] = kb[i];
    sV[r * LDSS_ + c] = vb[i];
  }
  __syncthreads();

  const int   count = (s_cnt < CAP_) ? s_cnt : CAP_;
  const int   nch   = (count + 15) >> 4;
  const float scale = 0.125f;  // 1/sqrt(dk=64)

  for (int ch = wave; ch < nch; ch += 2) {
    const int rows = ((count - ch * 16) < 16) ? (count - ch * 16) : 16;
    float* Qs = sQ[wave];
    float* Ps = sP[wave];

    // stage 16 gathered query rows (zero-fill padding rows)
    for (int i = lane; i < 16 * DK_; i += 32) {
      int r = i >> 6, c = i & 63;
      float val = 0.0f;
      if (r < rows) {
        int e = s_list[ch * 16 + r];
        val = qg[((size_t)h * TQ_ + (size_t)(e >> 1)) * DK_ + c];
      }
      Qs[r * LDSS_ + c] = val;
    }

    // ---- S[16q x 64k] = Q * K^T via V_WMMA_F32_16X16X4_F32 ----
    // A layout (16x4 f32): lane lm holds row M=lm; vgpr0/1 = K = 4s+2*lhalf +{0,1}
    // B layout (4x16 f32): lane lm holds col N=lm; vgpr0/1 = K = 4s+2*lhalf +{0,1}
    v8f acc[4];
    #pragma unroll
    for (int t = 0; t < 4; ++t) { v8f z = {0.f,0.f,0.f,0.f,0.f,0.f,0.f,0.f}; acc[t] = z; }
    {
      const float* Qb = &Qs[lm * LDSS_ + 2 * lhalf];
      for (int s = 0; s < 16; ++s) {
        v2f a; a.x = Qb[4 * s]; a.y = Qb[4 * s + 1];
        #pragma unroll
        for (int t = 0; t < 4; ++t) {
          const float* kp = &sK[(t * 16 + lm) * LDSS_ + 4 * s + 2 * lhalf];
          v2f b; b.x = kp[0]; b.y = kp[1];
          acc[t] = __builtin_amdgcn_wmma_f32_16x16x4_f32(
              false, a, false, b, (short)0, acc[t], false, false);
        }
      }
    }

    // ---- per-row partial softmax over this block's 64 keys ----
    // C/D layout: lane L, vgpr r -> S[q = r + 8*(L/16), key = 16t + L%16]
    #pragma unroll
    for (int r = 0; r < 8; ++r) {
      float s0 = acc[0][r] * scale, s1 = acc[1][r] * scale;
      float s2 = acc[2][r] * scale, s3 = acc[3][r] * scale;
      float mx = fmaxf(fmaxf(s0, s1), fmaxf(s2, s3));
      mx = fmaxf(mx, __shfl_xor(mx, 1, 32));
      mx = fmaxf(mx, __shfl_xor(mx, 2, 32));
      mx = fmaxf(mx, __shfl_xor(mx, 4, 32));
      mx = fmaxf(mx, __shfl_xor(mx, 8, 32));   // reduce within 16-lane half
      float p0 = __expf(s0 - mx), p1 = __expf(s1 - mx);
      float p2 = __expf(s2 - mx), p3 = __expf(s3 - mx);
      float ls = p0 + p1 + p2 + p3;
      ls += __shfl_xor(ls, 1, 32);
      ls += __shfl_xor(ls, 2, 32);
      ls += __shfl_xor(ls, 4, 32);
      ls += __shfl_xor(ls, 8, 32);
      int q_local = r + 8 * lhalf;
      if (lm == 0 && q_local < rows) {
        int e = s_list[ch * 16 + q_local];
        size_t idx = ((size_t)h * TQ_ + (size_t)(e >> 1)) * 2 + (size_t)(e & 1);
        pM[idx] = mx;
        pL[idx] = ls;
      }
      // stash P row (C-layout -> LDS) for A-layout reload in PV matmul
      Ps[q_local * LDSS_ +  0 + lm] = p0;
      Ps[q_local * LDSS_ + 16 + lm] = p1;
      Ps[q_local * LDSS_ + 32 + lm] = p2;
      Ps[q_local * LDSS_ + 48 + lm] = p3;
    }

    // ---- O_part[16q x 64d] = P * V via V_WMMA_F32_16X16X4_F32 ----
    v8f oacc[4];
    #pragma unroll
    for (int t = 0; t < 4; ++t) { v8f z = {0.f,0.f,0.f,0.f,0.f,0.f,0.f,0.f}; oacc[t] = z; }
    {
      const float* Pb = &Ps[lm * LDSS_ + 2 * lhalf];
      for (int s = 0; s < 16; ++s) {
        v2f a; a.x = Pb[4 * s]; a.y = Pb[4 * s + 1];
        #pragma unroll
        for (int t = 0; t < 4; ++t) {
          const float* vp = &sV[(4 * s + 2 * lhalf) * LDSS_ + t * 16 + lm];
          v2f b; b.x = vp[0]; b.y = vp[LDSS_];
          oacc[t] = __builtin_amdgcn_wmma_f32_16x16x4_f32(
              false, a, false, b, (short)0, oacc[t], false, false);
        }
      }
    }

    // ---- write per-(query,slot) partial output ----
    #pragma unroll
    for (int r = 0; r < 8; ++r) {
      int q_local = r + 8 * lhalf;
      if (q_local < rows) {
        int e = s_list[ch * 16 + q_local];
        size_t base =
            (((size_t)h * TQ_ + (size_t)(e >> 1)) * 2 + (size_t)(e & 1)) * DK_;
        pO[base +  0 + lm] = oacc[0][r];
        pO[base + 16 + lm] = oacc[1][r];
        pO[base + 32 + lm] = oacc[2][r];
        pO[base + 48 + lm] = oacc[3][r];
      }
    }
  }
}

// ---------------------------------------------------------------------------
// Kernel 2: merge the two per-slot partials per query (flash-attention combine).
// One thread per (h, q, d) output element.
// ---------------------------------------------------------------------------
__global__ __launch_bounds__(256) void bsattn_combine_kernel(
    const float* __restrict__ pO, const float* __restrict__ pM,
    const float* __restrict__ pL, float* __restrict__ out)
{
  int i = blockIdx.x * 256 + threadIdx.x;           // over H_*TQ_*DK_
  if (i >= H_ * TQ_ * DK_) return;
  int d  = i & (DK_ - 1);
  int hq = i >> 6;                                  // h*TQ_ + q
  float m0 = pM[(size_t)hq * 2 + 0], m1 = pM[(size_t)hq * 2 + 1];
  float l0 = pL[(size_t)hq * 2 + 0], l1 = pL[(size_t)hq * 2 + 1];
  float m  = fmaxf(m0, m1);
  float a0 = __expf(m0 - m);
  float a1 = __expf(m1 - m);
  float l  = l0 * a0 + l1 * a1;
  float o  = pO[((size_t)hq * 2 + 0) * DK_ + d] * a0 +
             pO[((size_t)hq * 2 + 1) * DK_ + d] * a1;
  out[i] = o / l;
}

// ---------------------------------------------------------------------------
// Host launcher
// ---------------------------------------------------------------------------
extern "C" void kernel_launch(void* const* d_in, const int* in_sizes, int n_in,
                              void* d_out, int out_size, void* d_ws, size_t ws_size,
                              hipStream_t stream) {
  const float* qg   = (const float*)d_in[0];
  const float* kg   = (const float*)d_in[1];
  const float* vg   = (const float*)d_in[2];
  // d_in[3] = BS scalar (compile-time 64 here)
  const int*   top2 = (const int*)d_in[4];

  // workspace: pO [H,TQ,2,DK] f32 | pM [H,TQ,2] | pL [H,TQ,2]
  float* pO = (float*)d_ws;
  float* pM = pO + (size_t)H_ * TQ_ * 2 * DK_;
  float* pL = pM + (size_t)H_ * TQ_ * 2;

  bsattn_partial_kernel<<<H_ * NB_, 64, 0, stream>>>(qg, kg, vg, top2, pO, pM, pL);

  int nelem = H_ * TQ_ * DK_;
  bsattn_combine_kernel<<<(nelem + 255) / 256, 256, 0, stream>>>(pO, pM, pL, (float*)d_out);
}